// Reconstructor_28518582845966
// MI455X (gfx1250) — compile-verified
//
#include <hip/hip_runtime.h>
#include <math.h>
#include <stdint.h>

// Problem constants (match reference)
#define DD 256            // box size
#define WR 129            // rfft width
#define WRP 144           // WR padded to multiple of 16 for WMMA tiling
#define BATCH 64
#define VOL (DD*DD*WR)    // 8,454,144 elements per output channel

#define PI2F 6.28318530717958647692f

// LDS A-tile row stride (floats): 260 -> banks (addr/4)%64 step 4 per row,
// so the 16 M-lanes of a ds_load_b64 hit 64 distinct banks (no conflicts),
// and rows stay 8-byte aligned.
#define LDSTRIDE 260

// Use gfx1250 async global->LDS copies for tile staging.
#define USE_ASYNC_LDS 1

typedef float v2f __attribute__((ext_vector_type(2)));
typedef float v8f __attribute__((ext_vector_type(8)));

// ---------------------------------------------------------------------------
// WMMA f32 16x16x4 fragment helpers (wave32).
// A (16x4, MxK): lanes 0-15 hold M=lane rows; VGPR0/1 hold K={0,1} for lanes
// 0-15 and K={2,3} for lanes 16-31 (ISA 7.12.2 "32-bit A-Matrix 16x4").
// B (4x16, KxN): mirrored — lanes hold N columns, VGPR x lane-half holds K.
// C/D (16x16): VGPR j -> M=j (lanes 0-15) / M=j+8 (lanes 16-31), N=lane&15.
// ---------------------------------------------------------------------------
__device__ __forceinline__ v2f ldA_lds(const float* __restrict__ lds,
                                       int k0, int lane) {
    int m = lane & 15;
    int k = k0 + ((lane >> 4) << 1);
    return *(const v2f*)&lds[m * LDSTRIDE + k];     // ds_load_b64
}

__device__ __forceinline__ v2f ldB(const float* __restrict__ B, int ldb,
                                   int k0, int n0, int lane) {
    int n = n0 + (lane & 15);
    int k = k0 + ((lane >> 4) << 1);
    v2f b; b.x = B[k * ldb + n]; b.y = B[(k + 1) * ldb + n];
    return b;
}

__device__ __forceinline__ void stC(float* __restrict__ C, int ldc,
                                    int m0, int n0, int lane, v8f acc) {
    int n  = n0 + (lane & 15);
    int mb = m0 + ((lane >> 4) << 3);
#pragma unroll
    for (int j = 0; j < 8; ++j) C[(mb + j) * ldc + n] = acc[j];
}

// ---------------------------------------------------------------------------
// Async global -> LDS 16-byte copy (CDNA5 GLOBAL_LOAD_ASYNC_TO_LDS_B128,
// tracked by ASYNCcnt). Fallback: plain load + LDS store.
// ---------------------------------------------------------------------------
__device__ __forceinline__ void cp_lds_b128(float* ldst, const float* gsrc) {
#if USE_ASYNC_LDS
    uint32_t loff = (uint32_t)(uintptr_t)(__attribute__((address_space(3))) float*)ldst;
    asm volatile("global_load_async_to_lds_b128 %0, %1, off"
                 :: "v"(loff), "v"((uint64_t)(uintptr_t)gsrc)
                 : "memory");
#else
    *(float4*)ldst = *(const float4*)gsrc;
#endif
}

__device__ __forceinline__ void stage_wait() {
#if USE_ASYNC_LDS
    asm volatile("s_wait_asynccnt 0x0" ::: "memory");
#endif
    __syncthreads();
}

// Stage one 16x256 f32 A-tile (contiguous 4096 floats in global, since
// lda == 256 == row length) into LDS with padded row stride.
// 1024 float4 tasks spread over 288 threads.
__device__ __forceinline__ void stage_tile(float* __restrict__ lds,
                                           const float* __restrict__ gsrc,
                                           int tid) {
    for (int t = tid; t < 1024; t += 288) {
        int row = t >> 6, chunk = t & 63;
        cp_lds_b128(&lds[row * LDSTRIDE + chunk * 4], gsrc + t * 4);
    }
}

// ---------------------------------------------------------------------------
// Zero the 4-channel output volume (harness poisons d_out with 0xAA).
// ---------------------------------------------------------------------------
__global__ void __launch_bounds__(256) zero_out(float4* __restrict__ out, int n4) {
    int i = blockIdx.x * blockDim.x + threadIdx.x;
    if (i < n4) out[i] = make_float4(0.f, 0.f, 0.f, 0.f);
}

// ---------------------------------------------------------------------------
// Twiddle matrices.
// Cx[x,k] = cos(2*pi*x*k/D), Sxn[x,k] = -sin(...) for k<WR, zero-padded to WRP.
// (numpy rfft: X[k] = sum_x x[x] * exp(-2pi i x k / D))
// ---------------------------------------------------------------------------
__global__ void __launch_bounds__(256) twiddle_x(float* __restrict__ Cx,
                                                 float* __restrict__ Sxn) {
    int i = blockIdx.x * blockDim.x + threadIdx.x;
    if (i >= DD * WRP) return;
    int x = i / WRP, k = i % WRP;
    float c = 0.f, s = 0.f;
    if (k < WR) {
        int t = (x * k) & (DD - 1);            // exact argument reduction
        __sincosf((float)t * (PI2F / (float)DD), &s, &c);
    }
    Cx[i] = c; Sxn[i] = -s;
}

// Cy[r,y] = cos(2*pi*y*(r-128)/D); Syp = +sin; Syn = -sin.
// Output row r corresponds to fftshifted frequency ky = r - D/2, so the
// fftshift is folded directly into the DFT matrix.
__global__ void __launch_bounds__(256) twiddle_y(float* __restrict__ Cy,
                                                 float* __restrict__ Syp,
                                                 float* __restrict__ Syn) {
    int i = blockIdx.x * blockDim.x + threadIdx.x;  // over D*D
    int r = i >> 8, y = i & (DD - 1);
    int t = (y * (r - DD / 2)) & (DD - 1);          // two's-complement & == mod 256
    float s, c;
    __sincosf((float)t * (PI2F / (float)DD), &s, &c);
    Cy[i] = c; Syp[i] = s; Syn[i] = -s;
}

// ---------------------------------------------------------------------------
// GEMM1: G = rfft along x.  Gre = img @ Cx, Gim = img @ Sxn.
// Block = 288 threads (9 waves) = one (b, mt) tile-row; the shared 16x256
// A-tile of the image is staged once into LDS via async copies; each wave
// computes one 16x16 output tile (nt = wave id). 2 WMMAs per k-step.
// ---------------------------------------------------------------------------
__global__ void __launch_bounds__(288) gemm_rfft_x(const float* __restrict__ imgs,
                                                   const float* __restrict__ Cx,
                                                   const float* __restrict__ Sxn,
                                                   float* __restrict__ Gre,
                                                   float* __restrict__ Gim) {
    __shared__ float ldsA[16 * LDSTRIDE];

    const int lane = threadIdx.x & 31;
    const int nt   = threadIdx.x >> 5;          // 0..8
    const int mt   = blockIdx.x;                // 0..15
    const int b    = blockIdx.y;                // 0..63
    const int m0 = mt * 16, n0 = nt * 16;

    stage_tile(ldsA, imgs + (size_t)b * DD * DD + (size_t)m0 * DD, threadIdx.x);
    stage_wait();

    v8f acc_re = {}; v8f acc_im = {};
#pragma unroll 4
    for (int k0 = 0; k0 < DD; k0 += 4) {
        v2f a  = ldA_lds(ldsA, k0, lane);
        v2f bc = ldB(Cx,  WRP, k0, n0, lane);
        v2f bs = ldB(Sxn, WRP, k0, n0, lane);
        acc_re = __builtin_amdgcn_wmma_f32_16x16x4_f32(false, a, false, bc,
                                                       (short)0, acc_re, false, false);
        acc_im = __builtin_amdgcn_wmma_f32_16x16x4_f32(false, a, false, bs,
                                                       (short)0, acc_im, false, false);
    }
    stC(Gre + (size_t)b * DD * WRP, WRP, m0, n0, lane, acc_re);
    stC(Gim + (size_t)b * DD * WRP, WRP, m0, n0, lane, acc_im);
}

// ---------------------------------------------------------------------------
// GEMM2: F = (fftshifted) complex DFT along y.
// With twiddle T = Cy + i*Syn (Syn = -sin):
//   Fre = Cy@Gre + Syp@Gim        (Syp = +sin)
//   Fim = Cy@Gim + Syn@Gre
// Three 16x256 A-tiles (Cy/Syp/Syn rows m0..m0+15) staged in LDS (~50 KB),
// 4 pure-accumulation WMMAs per k-step.
// ---------------------------------------------------------------------------
__global__ void __launch_bounds__(288) gemm_fft_y(const float* __restrict__ Gre,
                                                  const float* __restrict__ Gim,
                                                  const float* __restrict__ Cy,
                                                  const float* __restrict__ Syp,
                                                  const float* __restrict__ Syn,
                                                  float* __restrict__ Fre,
                                                  float* __restrict__ Fim) {
    __shared__ float ldsC[16 * LDSTRIDE];
    __shared__ float ldsP[16 * LDSTRIDE];
    __shared__ float ldsN[16 * LDSTRIDE];

    const int lane = threadIdx.x & 31;
    const int nt   = threadIdx.x >> 5;          // 0..8
    const int mt   = blockIdx.x;                // 0..15
    const int b    = blockIdx.y;                // 0..63
    const int m0 = mt * 16, n0 = nt * 16;

    stage_tile(ldsC, Cy  + (size_t)m0 * DD, threadIdx.x);
    stage_tile(ldsP, Syp + (size_t)m0 * DD, threadIdx.x);
    stage_tile(ldsN, Syn + (size_t)m0 * DD, threadIdx.x);
    stage_wait();

    const float* Br = Gre + (size_t)b * DD * WRP;
    const float* Bi = Gim + (size_t)b * DD * WRP;

    v8f acc_re = {}; v8f acc_im = {};
#pragma unroll 2
    for (int k0 = 0; k0 < DD; k0 += 4) {
        v2f ac  = ldA_lds(ldsC, k0, lane);
        v2f ap  = ldA_lds(ldsP, k0, lane);
        v2f an  = ldA_lds(ldsN, k0, lane);
        v2f bre = ldB(Br, WRP, k0, n0, lane);
        v2f bim = ldB(Bi, WRP, k0, n0, lane);
        acc_re = __builtin_amdgcn_wmma_f32_16x16x4_f32(false, ac, false, bre,
                                                       (short)0, acc_re, false, false);
        acc_re = __builtin_amdgcn_wmma_f32_16x16x4_f32(false, ap, false, bim,
                                                       (short)0, acc_re, false, false);
        acc_im = __builtin_amdgcn_wmma_f32_16x16x4_f32(false, ac, false, bim,
                                                       (short)0, acc_im, false, false);
        acc_im = __builtin_amdgcn_wmma_f32_16x16x4_f32(false, an, false, bre,
                                                       (short)0, acc_im, false, false);
    }
    stC(Fre + (size_t)b * DD * WRP, WRP, m0, n0, lane, acc_re);
    stC(Fim + (size_t)b * DD * WRP, WRP, m0, n0, lane, acc_im);
}

// ---------------------------------------------------------------------------
// Fused phase-shift + CTF + rotate + Hermitian fold + trilinear atomic scatter.
// One thread per (b, r, kx). The 135 MB output volume is L2-resident on
// MI455X (192 MB L2), so the ~68M atomic f32 adds never touch HBM.
// ---------------------------------------------------------------------------
__global__ void __launch_bounds__(160) scatter_bp(const float* __restrict__ Fre,
                                                  const float* __restrict__ Fim,
                                                  const float* __restrict__ ctf,
                                                  const float* __restrict__ rotMats,
                                                  const float* __restrict__ shifts,
                                                  float* __restrict__ out) {
    const int kx = threadIdx.x;
    if (kx >= WR) return;
    const int r = blockIdx.y;
    const int b = blockIdx.z;

    const float ky  = (float)(r - DD / 2);
    const float fkx = (float)kx;
    const float* R  = rotMats + b * 9;
    const float sy = shifts[b * 2 + 0];
    const float sx = shifts[b * 2 + 1];

    float fre = Fre[((size_t)b * DD + r) * WRP + kx];
    float fim = Fim[((size_t)b * DD + r) * WRP + kx];
    float c   = ctf[((size_t)b * DD + r) * WR + kx];

    // phase = exp(-2*pi*i*(ky*sy + kx*sx)/D)
    float ang = (-PI2F / (float)DD) * (ky * sy + fkx * sx);
    float ps, pc; __sincosf(ang, &ps, &pc);
    float vr = c * (fre * pc - fim * ps);
    float vi = c * (fre * ps + fim * pc);

    // rot = R @ [0, ky, kx]
    float rz = R[1] * ky + R[2] * fkx;
    float ry = R[4] * ky + R[5] * fkx;
    float rx = R[7] * ky + R[8] * fkx;
    if (rx < 0.f) { rz = -rz; ry = -ry; rx = -rx; vi = -vi; }  // Hermitian fold

    const float csq = c * c;
    float zc = rz + (float)(DD / 2);
    float yc = ry + (float)(DD / 2);
    float xc = rx;
    float z0f = floorf(zc), y0f = floorf(yc), x0f = floorf(xc);
    float fz = zc - z0f, fy = yc - y0f, fx = xc - x0f;
    int z0 = (int)z0f, y0 = (int)y0f, x0 = (int)x0f;

#pragma unroll
    for (int dz = 0; dz < 2; ++dz) {
        int zi = z0 + dz;
        if (zi < 0 || zi >= DD) continue;
        float wz = dz ? fz : 1.f - fz;
#pragma unroll
        for (int dy = 0; dy < 2; ++dy) {
            int yi = y0 + dy;
            if (yi < 0 || yi >= DD) continue;
            float wzy = wz * (dy ? fy : 1.f - fy);
#pragma unroll
            for (int dx = 0; dx < 2; ++dx) {
                int xi = x0 + dx;
                if (xi < 0 || xi >= WR) continue;
                float w = wzy * (dx ? fx : 1.f - fx);
                int idx = (zi * DD + yi) * WR + xi;
                atomicAdd(out + idx,            w * vr);
                atomicAdd(out + VOL + idx,      w * vi);
                atomicAdd(out + 2 * VOL + idx,  w);
                atomicAdd(out + 3 * VOL + idx,  w * csq);
            }
        }
    }
}

// ---------------------------------------------------------------------------
extern "C" void kernel_launch(void* const* d_in, const int* in_sizes, int n_in,
                              void* d_out, int out_size, void* d_ws, size_t ws_size,
                              hipStream_t stream) {
    const float* imgs    = (const float*)d_in[0];   // [B, D, D]
    const float* ctf     = (const float*)d_in[1];   // [B, D, WR]
    const float* rotMats = (const float*)d_in[2];   // [B, 3, 3]
    const float* shifts  = (const float*)d_in[3];   // [B, 2]
    float* out = (float*)d_out;                     // [4, D, D, WR]

    // Workspace carve-up (~39 MB)
    float* ws  = (float*)d_ws;
    size_t o = 0;
    float* Cx  = ws + o; o += (size_t)DD * WRP;
    float* Sxn = ws + o; o += (size_t)DD * WRP;
    float* Cy  = ws + o; o += (size_t)DD * DD;
    float* Syp = ws + o; o += (size_t)DD * DD;
    float* Syn = ws + o; o += (size_t)DD * DD;
    float* Gre = ws + o; o += (size_t)BATCH * DD * WRP;
    float* Gim = ws + o; o += (size_t)BATCH * DD * WRP;
    float* Fre = ws + o; o += (size_t)BATCH * DD * WRP;
    float* Fim = ws + o; o += (size_t)BATCH * DD * WRP;

    // 1) zero the output accumulators
    zero_out<<<VOL / 256, 256, 0, stream>>>((float4*)out, VOL);

    // 2) twiddle matrices
    twiddle_x<<<(DD * WRP + 255) / 256, 256, 0, stream>>>(Cx, Sxn);
    twiddle_y<<<(DD * DD) / 256, 256, 0, stream>>>(Cy, Syp, Syn);

    // 3) rfft along x (real GEMM on fp32 WMMA, async-LDS staged A tiles)
    dim3 gg(DD / 16, BATCH);                 // (16, 64)
    gemm_rfft_x<<<gg, 288, 0, stream>>>(imgs, Cx, Sxn, Gre, Gim);

    // 4) fftshifted complex DFT along y
    gemm_fft_y<<<gg, 288, 0, stream>>>(Gre, Gim, Cy, Syp, Syn, Fre, Fim);

    // 5) phase/ctf/rotate/fold/trilinear atomic scatter
    dim3 gs(1, DD, BATCH);
    scatter_bp<<<gs, 160, 0, stream>>>(Fre, Fim, ctf, rotMats, shifts, out);
}